// EquivariantMatmulKernel_38225208934540
// MI455X (gfx1250) — compile-verified
//
#include <hip/hip_runtime.h>

// CDNA5 / gfx1250: wave32, WMMA f32 16x16x4 (K=4 == D_IN)
typedef __attribute__((ext_vector_type(2))) float v2f;
typedef __attribute__((ext_vector_type(8))) float v8f;

// Problem shapes from the reference:
//   basis         [E, 4, 4]   fp32   (streamed once -> NT)
//   edge_weights  [E, 2, 32]  fp32   (streamed once -> NT)
//   node_features [N, 4, 32]  fp32   (25.6 MB, reused ~16x -> keep in L2, RT)
//   U             [E]         int32
//   out           [E, 4, 32]  fp32   (streamed once -> NT stores)
__global__ __launch_bounds__(256) void eqmm_wmma_kernel(
    const float* __restrict__ basis,
    const float* __restrict__ ew,
    const float* __restrict__ nf,
    const int*   __restrict__ U,
    float*       __restrict__ out,
    int E)
{
    // One wave (32 lanes) per edge; 8 waves per 256-thread block.
    int e = __builtin_amdgcn_readfirstlane(
        (int)(blockIdx.x * 8u + (threadIdx.x >> 5)));
    if (e >= E) return;                       // wave-uniform branch (EXEC stays all-1s)

    const int lane = threadIdx.x & 31;
    const int rmod = lane & 15;               // A-row / B-column within half-wave
    const int half = lane >> 4;               // selects K pair {0,1} vs {2,3}

    // ---- Build A operands (16x4 f32 layout: lane L holds A[L&15][half*2 + {0,1}]) ----
    // A1: rows 0-3  = basis(e), rest 0  -> C rows 0-3  (lanes 0-15, channels 0-15)
    // A2: rows 8-11 = basis(e), rest 0  -> C rows 8-11 (lanes 16-31, channels 16-31)
    const bool a1 = (rmod < 4);
    const bool a2 = (rmod >= 8) && (rmod < 12);
    const int  r  = a1 ? rmod : (rmod - 8);

    v2f bv = {0.0f, 0.0f};
    if (a1 || a2) {
        const float* bp = basis + (size_t)e * 16 + r * 4 + half * 2;
        bv.x = __builtin_nontemporal_load(bp + 0);   // streamed once: TH_NT
        bv.y = __builtin_nontemporal_load(bp + 1);
    }
    v2f A1 = a1 ? bv : (v2f){0.0f, 0.0f};
    v2f A2 = a2 ? bv : (v2f){0.0f, 0.0f};

    // ---- Gather source-node features, build B operands (4x16 f32 layout) ----
    // B[k][n]: lane n holds B[half*2][n] in v0, B[half*2+1][n] in v1.
    // Regular-temporal: the 25.6 MB table is the only reused data, keep it in L2.
    const int u = __builtin_amdgcn_readfirstlane(U[e]);
    const float* xp = nf + (size_t)u * 128 + half * 64 + rmod;  // x[half*2][rmod]
    v2f B1, B2;
    B1.x = xp[0];        // x[half*2    ][rmod]        (channels 0-15)
    B1.y = xp[32];       // x[half*2 + 1][rmod]
    B2.x = xp[16];       // x[half*2    ][16 + rmod]   (channels 16-31)
    B2.y = xp[48];       // x[half*2 + 1][16 + rmod]

    // ---- Two accumulating WMMAs fill all 32 lanes' d[0..3] with out[o][lane] ----
    v8f d = {};
    d = __builtin_amdgcn_wmma_f32_16x16x4_f32(
            false, A1, false, B1, (short)0, d, false, false);
    d = __builtin_amdgcn_wmma_f32_16x16x4_f32(
            false, A2, false, B2, (short)0, d, false, false);

    // ---- Degree-wise radial weights (streamed once: TH_NT) ----
    const float* wp = ew + (size_t)e * 64 + lane;
    const float w0 = __builtin_nontemporal_load(wp + 0);    // degree 0 (row 0)
    const float w1 = __builtin_nontemporal_load(wp + 32);   // degree 1 (rows 1-3)

    // ---- Store: 4 full-wave 128B coalesced non-temporal B32 stores ----
    float* op = out + (size_t)e * 128 + lane;
    __builtin_nontemporal_store(d[0] * w0, op + 0);
    __builtin_nontemporal_store(d[1] * w1, op + 32);
    __builtin_nontemporal_store(d[2] * w1, op + 64);
    __builtin_nontemporal_store(d[3] * w1, op + 96);
}

extern "C" void kernel_launch(void* const* d_in, const int* in_sizes, int n_in,
                              void* d_out, int out_size, void* d_ws, size_t ws_size,
                              hipStream_t stream) {
    const float* basis = (const float*)d_in[0];   // [E,4,4]
    const float* ew    = (const float*)d_in[1];   // [E,2,32]
    const float* nf    = (const float*)d_in[2];   // [N,4,32]
    const int*   U     = (const int*)d_in[3];     // [E]
    float* out = (float*)d_out;                   // [E,4,32]

    const int E = in_sizes[3];
    const int edges_per_block = 8;                // 8 wave32 waves / 256 threads
    dim3 grid((E + edges_per_block - 1) / edges_per_block);
    eqmm_wmma_kernel<<<grid, 256, 0, stream>>>(basis, ew, nf, U, out, E);
}